// NestedGCN_4887672783292
// MI455X (gfx1250) — compile-verified
//
#include <hip/hip_runtime.h>
#include <hip/hip_bf16.h>
#include <math.h>

typedef __attribute__((ext_vector_type(2))) float v2f;
typedef __attribute__((ext_vector_type(8))) float v8f;

#define HID 128

// ---------------- utility kernels ----------------

__global__ void fill_kernel(float* __restrict__ p, float v, int n) {
    int t = blockIdx.x * blockDim.x + threadIdx.x;
    if (t < n) p[t] = v;
}

__global__ void deg_count_kernel(float* __restrict__ deg, const int* __restrict__ dst, int E) {
    int t = blockIdx.x * blockDim.x + threadIdx.x;
    if (t < E) atomicAdd(&deg[dst[t]], 1.0f);
}

__global__ void rsqrt_kernel(float* __restrict__ d, int n) {
    int t = blockIdx.x * blockDim.x + threadIdx.x;
    if (t < n) {
        float v = d[t];
        d[t] = (v > 0.f) ? rsqrtf(v) : 0.f;
    }
}

// ---------------- layer 1: Z = x@W1 ; X = Z*dinv^2 + b1 (fused) ----------------

__global__ void gemm_in2_kernel(const float* __restrict__ x, const float* __restrict__ W1,
                                const float* __restrict__ dinv, const float* __restrict__ bias,
                                float* __restrict__ Z, float* __restrict__ X, int N) {
    int t = blockIdx.x * blockDim.x + threadIdx.x;
    if (t < N * HID) {
        int n = t >> 7, f = t & (HID - 1);
        float v = x[n * 2 + 0] * W1[f] + x[n * 2 + 1] * W1[HID + f];
        Z[t] = v;
        float dv = dinv[n];
        X[t] = v * dv * dv + bias[f];
    }
}

// ---------------- WMMA GEMM: Z[N,128] = A[N,128] @ W[128,128] (fp32) ----------------
// Block = 256 threads = 8 waves; each wave computes a 16-row strip x all 128 cols.
// W staged in LDS (64 KB) in FRAGMENT ORDER: element W[k][n] lives at word index
//   (k>>1)*256 + 2*n + (k&1)
// so the (W[k][n], W[k+1][n]) pair a lane needs is one aligned ds_load_b64 that
// lands directly in the even VGPR pair v_wmma_f32_16x16x4_f32 consumes.
// Epilogue also writes X = Z*dinv[row]^2 + bias[col] (self-loop + bias fused).

__global__ void __launch_bounds__(256) gemm128_wmma_kernel(const float* __restrict__ A,
                                                           const float* __restrict__ W,
                                                           const float* __restrict__ dinv,
                                                           const float* __restrict__ bias,
                                                           float* __restrict__ Out,
                                                           float* __restrict__ Xout,
                                                           int nrows) {
    __shared__ float ldsW[HID * HID];   // 64 KB, fragment-order layout (see above)
    const int tid = threadIdx.x;
    for (int i = tid; i < HID * HID; i += 256) {
        int k = i >> 7, n = i & (HID - 1);
        ldsW[((k >> 1) << 8) + (n << 1) + (k & 1)] = W[i];
    }
    __syncthreads();

    const int lane = tid & 31;
    const int wave = tid >> 5;
    const int m    = lane & 15;
    const int kh   = (lane >> 4) << 1;          // 0 for lanes 0-15, 2 for lanes 16-31
    const int rowBase = blockIdx.x * 128 + wave * 16;
    // OOB A rows only feed OOB D rows (never stored) -> clamp address, load unconditionally
    int rowc = rowBase + m;
    if (rowc > nrows - 1) rowc = nrows - 1;
    const float* __restrict__ arow = A + (size_t)rowc * HID + kh;

    const int ln2 = (lane & 15) << 1;

    v8f acc[8] = {};

    for (int k0 = 0; k0 < HID; k0 += 4) {
        // A fragment: A[m][k0+kh], A[m][k0+kh+1] -> one global_load_b64
        v2f a = *(const v2f*)(arow + k0);
        // B fragments: one ds_load_b64 each, already packed (W[k][n], W[k+1][n])
        const float* __restrict__ wp = &ldsW[(((k0 + kh) >> 1) << 8) + ln2];
#pragma unroll
        for (int nt = 0; nt < 8; ++nt) {
            v2f b = *(const v2f*)(wp + nt * 32);
            acc[nt] = __builtin_amdgcn_wmma_f32_16x16x4_f32(
                false, a, false, b, (short)0, acc[nt], false, false);
        }
    }

    // C/D layout: VGPR r -> row (r + 8*(lane>=16)), col = n0 + (lane&15)
    const int orow0 = rowBase + ((lane >> 4) << 3);
    const int col0  = lane & 15;
    float bcol[8];
#pragma unroll
    for (int nt = 0; nt < 8; ++nt) bcol[nt] = bias[nt * 16 + col0];

#pragma unroll
    for (int r = 0; r < 8; ++r) {
        int orow = orow0 + r;
        if (orow < nrows) {
            float dv = dinv[orow];
            float s  = dv * dv;
            size_t base = (size_t)orow * HID + col0;
#pragma unroll
            for (int nt = 0; nt < 8; ++nt) {
                float v = acc[nt][r];
                Out[base + nt * 16]  = v;
                Xout[base + nt * 16] = v * s + bcol[nt];
            }
        }
    }
}

// ---------------- edge aggregation ----------------
// X[dst] += Z[src] * dinv[src]*dinv[dst] ; 32 lanes per edge, float4 per lane

__global__ void agg_edges_kernel(float* __restrict__ X, const float* __restrict__ Z,
                                 const float* __restrict__ dinv,
                                 const int* __restrict__ src, const int* __restrict__ dst,
                                 int E) {
    int t = blockIdx.x * blockDim.x + threadIdx.x;
    int e = t >> 5;
    if (e >= E) return;
    int j = t & 31;
    int s = src[e], d = dst[e];
    float w = dinv[s] * dinv[d];
    const float4 hv = *(const float4*)(Z + (size_t)s * HID + j * 4);
    float* o = X + (size_t)d * HID + j * 4;
    atomicAdd(o + 0, hv.x * w);
    atomicAdd(o + 1, hv.y * w);
    atomicAdd(o + 2, hv.z * w);
    atomicAdd(o + 3, hv.w * w);
}

// ---------------- pooling: nodes -> graphs (composed segment sums) ----------------

__global__ void pool_scatter_kernel(float* __restrict__ pooled, const float* __restrict__ X,
                                    const int* __restrict__ n2s, const int* __restrict__ s2g,
                                    int N) {
    int t = blockIdx.x * blockDim.x + threadIdx.x;
    int n = t >> 5;
    if (n >= N) return;
    int j = t & 31;
    int g = s2g[n2s[n]];
    const float4 hv = *(const float4*)(X + (size_t)n * HID + j * 4);
    float* o = pooled + (size_t)g * HID + j * 4;
    atomicAdd(o + 0, hv.x);
    atomicAdd(o + 1, hv.y);
    atomicAdd(o + 2, hv.z);
    atomicAdd(o + 3, hv.w);
}

// ---------------- head: relu(pooled@W1+b1)@W2+b2 -> log_softmax ----------------

__global__ void head_kernel(const float* __restrict__ pooled,
                            const float* __restrict__ W1, const float* __restrict__ b1,
                            const float* __restrict__ W2, const float* __restrict__ b2,
                            float* __restrict__ out) {
    __shared__ float sh[64 * HID];   // 32 KB
    int t = threadIdx.x;
    for (int i = t; i < 64 * HID; i += 256) sh[i] = pooled[i];
    __syncthreads();
    if (t < 64) {
        float l0 = b2[0], l1 = b2[1];
        for (int f = 0; f < HID; ++f) {
            float acc = b1[f];
            for (int k = 0; k < HID; ++k) acc += sh[t * HID + k] * W1[k * HID + f];
            float h = acc > 0.f ? acc : 0.f;
            l0 += h * W2[f * 2 + 0];
            l1 += h * W2[f * 2 + 1];
        }
        float m = fmaxf(l0, l1);
        float lse = m + logf(expf(l0 - m) + expf(l1 - m));
        out[t * 2 + 0] = l0 - lse;
        out[t * 2 + 1] = l1 - lse;
    }
}

// ---------------- launch ----------------

extern "C" void kernel_launch(void* const* d_in, const int* in_sizes, int n_in,
                              void* d_out, int out_size, void* d_ws, size_t ws_size,
                              hipStream_t stream) {
    (void)n_in; (void)out_size; (void)ws_size;

    const float* x    = (const float*)d_in[0];
    const int*   eidx = (const int*)d_in[1];
    const int*   n2s  = (const int*)d_in[2];
    const int*   s2g  = (const int*)d_in[3];
    const float* W1   = (const float*)d_in[4];
    const float* b1   = (const float*)d_in[5];
    const float* Ws   = (const float*)d_in[6];
    const float* bs   = (const float*)d_in[7];
    const float* l1W  = (const float*)d_in[8];
    const float* l1b  = (const float*)d_in[9];
    const float* l2W  = (const float*)d_in[10];
    const float* l2b  = (const float*)d_in[11];
    float* out = (float*)d_out;

    const int N = in_sizes[0] / 2;      // 100000
    const int E = in_sizes[1] / 2;      // 1600000
    const int* src = eidx;
    const int* dst = eidx + E;

    float* ws = (float*)d_ws;
    size_t off = ((size_t)N + 255) & ~(size_t)255;
    float* dinv   = ws;                           // N
    float* X      = ws + off;                     // N*128
    float* Z      = X + (size_t)N * HID;          // N*128
    float* pooled = Z + (size_t)N * HID;          // 64*128

    const int thr = 256;
    const int gN   = (N + thr - 1) / thr;
    const int gE   = (E + thr - 1) / thr;
    const int gNH  = (N * HID + thr - 1) / thr;
    const int gE32 = (int)(((size_t)E * 32 + thr - 1) / thr);
    const int gN32 = (int)(((size_t)N * 32 + thr - 1) / thr);

    // degrees & symmetric norm (self-loop contributes the initial 1.0)
    fill_kernel<<<gN, thr, 0, stream>>>(dinv, 1.0f, N);
    deg_count_kernel<<<gE, thr, 0, stream>>>(dinv, dst, E);
    rsqrt_kernel<<<gN, thr, 0, stream>>>(dinv, N);

    // layer 1 (Fin=2), fused self-loop+bias epilogue
    gemm_in2_kernel<<<gNH, thr, 0, stream>>>(x, W1, dinv, b1, Z, X, N);
    agg_edges_kernel<<<gE32, thr, 0, stream>>>(X, Z, dinv, src, dst, E);

    // layers 2..4 (128x128 via WMMA), fused self-loop+bias epilogue
    const int gGemm = (N + 127) / 128;
    for (int l = 0; l < 3; ++l) {
        gemm128_wmma_kernel<<<gGemm, 256, 0, stream>>>(
            X, Ws + (size_t)l * HID * HID, dinv, bs + (size_t)l * HID, Z, X, N);
        agg_edges_kernel<<<gE32, thr, 0, stream>>>(X, Z, dinv, src, dst, E);
    }

    // pooling (node -> subgraph -> graph composes to node -> graph)
    fill_kernel<<<(64 * HID + thr - 1) / thr, thr, 0, stream>>>(pooled, 0.0f, 64 * HID);
    pool_scatter_kernel<<<gN32, thr, 0, stream>>>(pooled, X, n2s, s2g, N);

    // MLP head + log_softmax
    head_kernel<<<1, 256, 0, stream>>>(pooled, l1W, l1b, l2W, l2b, out);
}